// RiemannianMetric_83923660964339
// MI455X (gfx1250) — compile-verified
//
#include <hip/hip_runtime.h>
#include <hip/hip_bf16.h>

// Problem constants (B=2, S=1024, DIM=256, RANK=32)
#define NPOS  2048      // B*S
#define DIMN  256
#define RANKN 32
#define LSTR  34        // LDS row stride (floats): 32 distinct banks, keeps 8B align

typedef __attribute__((ext_vector_type(2))) float v2f;
typedef __attribute__((ext_vector_type(8))) float v8f;

// Stage 1: m2[bs, r] = tanh(x[bs,:]·W[r,:] + b[r])^2
__global__ void __launch_bounds__(256)
rm_mod2_kernel(const float* __restrict__ x, const float* __restrict__ W,
               const float* __restrict__ bvec, float* __restrict__ m2) {
  int idx = blockIdx.x * 256 + threadIdx.x;   // 0 .. NPOS*RANKN-1
  int r   = idx & (RANKN - 1);
  int bs  = idx >> 5;
  const float4* xp = (const float4*)(x + (size_t)bs * DIMN);
  const float4* wp = (const float4*)(W + (size_t)r * DIMN);
  float acc = bvec[r];
#pragma unroll 8
  for (int d = 0; d < DIMN / 4; ++d) {
    float4 xv = xp[d];
    float4 wv = wp[d];
    acc += xv.x * wv.x + xv.y * wv.y + xv.z * wv.z + xv.w * wv.w;
  }
  float t = tanhf(acc);
  m2[idx] = t * t;
}

// Stage 2: per position bs, g = A^T diag(m2) A + lam*I  (256x256 f32 tile)
// One 256-thread block (8 wave32) per position; fp32 WMMA 16x16x4, K=32.
__global__ void __launch_bounds__(256)
rm_gram_kernel(const float* __restrict__ A, const float* __restrict__ m2,
               const float* __restrict__ log_lambda, float* __restrict__ out) {
  __shared__ float At[DIMN * LSTR];   // A transposed: At[i*LSTR + r] = A[r, i]
  __shared__ float sws[RANKN];        // per-position mod^2 weights

  const int bs  = blockIdx.x;
  const int tid = threadIdx.x;
  const float lam = __expf(log_lambda[0]);

  if (tid < RANKN) sws[tid] = m2[(size_t)bs * RANKN + tid];
  // stage A^T: thread tid owns column i = tid; coalesced global reads per r
#pragma unroll
  for (int r = 0; r < RANKN; ++r)
    At[tid * LSTR + r] = A[(size_t)r * DIMN + tid];
  __syncthreads();

  const int wave = tid >> 5;
  const int lane = tid & 31;
  const int half = lane >> 4;     // 0: lanes 0-15, 1: lanes 16-31
  const int mrow = lane & 15;     // M (for A op) / N (for B op) within tile
  const int koff = half * 2;      // K offset per WMMA f32 16x16x4 layout

  float* __restrict__ outp = out + (size_t)bs * DIMN * DIMN;

  for (int Ii = 0; Ii < 2; ++Ii) {
    const int I = wave + Ii * 8;          // tile row, 0..15
    // A-operand for this tile row: Atilde[m, k] = s[r] * A[r, I*16+m]
    v2f a[8];
#pragma unroll
    for (int c = 0; c < 8; ++c) {
      v2f av = *(const v2f*)&At[(I * 16 + mrow) * LSTR + c * 4 + koff];
      av.x *= sws[c * 4 + koff];
      av.y *= sws[c * 4 + koff + 1];
      a[c] = av;
    }
    for (int J = 0; J < 16; ++J) {        // tile col, 0..15
      v8f acc = {};
#pragma unroll
      for (int c = 0; c < 8; ++c) {
        // B-operand: B[k, n] = A[r, J*16+n]  (raw A^T from LDS)
        v2f bv = *(const v2f*)&At[(J * 16 + mrow) * LSTR + c * 4 + koff];
        acc = __builtin_amdgcn_wmma_f32_16x16x4_f32(
            /*neg_a=*/false, a[c], /*neg_b=*/false, bv,
            /*c_mod=*/(short)0, acc, /*reuse_a=*/false, /*reuse_b=*/false);
      }
      // Diagonal lam fixup only possible in the diagonal tile; I and J are
      // wave-uniform, so this is a scalar branch (no per-lane cndmask chain
      // in the 15/16 off-diagonal iterations).
      if (I == J) {
#pragma unroll
        for (int v = 0; v < 8; ++v)
          if (v + half * 8 == mrow) acc[v] += lam;
      }
      const int j = J * 16 + mrow;
#pragma unroll
      for (int v = 0; v < 8; ++v) {
        const int i = I * 16 + v + half * 8;
        // 512 MiB streaming output >> 192 MB L2: non-temporal store
        __builtin_nontemporal_store(acc[v], &outp[(size_t)i * DIMN + j]);
      }
    }
  }
}

extern "C" void kernel_launch(void* const* d_in, const int* in_sizes, int n_in,
                              void* d_out, int out_size, void* d_ws, size_t ws_size,
                              hipStream_t stream) {
  const float* x          = (const float*)d_in[0];  // [2,1024,256]
  const float* A          = (const float*)d_in[1];  // [32,256]
  const float* log_lambda = (const float*)d_in[2];  // scalar
  const float* W          = (const float*)d_in[3];  // [32,256]
  const float* b          = (const float*)d_in[4];  // [32]
  float* out = (float*)d_out;                       // [2,1024,256,256]
  float* m2  = (float*)d_ws;                        // NPOS*RANKN floats (256 KiB)

  rm_mod2_kernel<<<(NPOS * RANKN) / 256, 256, 0, stream>>>(x, W, b, m2);
  rm_gram_kernel<<<NPOS, 256, 0, stream>>>(A, m2, log_lambda, out);
}